// Tokenizer_32813550141817
// MI455X (gfx1250) — compile-verified
//
#include <hip/hip_runtime.h>
#include <hip/hip_bf16.h>

// VQ tokenizer: fused dist-GEMM + argmin + gather + commitment loss.
// Shapes: z (16,2048,256) f32, codebook (2048,256) f32, mask (16,2048,1) f32.
// M = 32768 rows, K = 256, N = 2048 codes.
//
// argmin_v ||z - e_v||^2 == argmax_v ( 2*z.e_v - ||e_v||^2 ), so the distance
// matrix (256 MB) is never materialized. fp32 WMMA (V_WMMA_F32_16X16X4_F32)
// keeps argmin faithful to the fp32 reference while using the matrix pipes.
//
// v2: codebook tiles stream through LDS with double-buffered
// GLOBAL_LOAD_ASYNC_TO_LDS_B64 (ASYNCcnt-tracked), so the per-tile global
// latency is hidden behind the 64 WMMAs of the previous tile.

typedef __attribute__((ext_vector_type(2))) float v2f;
typedef __attribute__((ext_vector_type(8))) float v8f;
typedef __attribute__((ext_vector_type(2))) int   v2i;

#define C_DIM 256
#define V_DIM 2048
#define M_DIM 32768   // B*T = 16*2048
#define NTILES (V_DIM / 16)

#if defined(__has_builtin)
# if __has_builtin(__builtin_amdgcn_global_load_async_to_lds_b64) && \
     __has_builtin(__builtin_amdgcn_s_wait_asynccnt)
#  define VQ_ASYNC 1
# endif
#endif
#ifndef VQ_ASYNC
# define VQ_ASYNC 0
#endif

// ---------------------------------------------------------------- e-norms
__global__ void vq_enorm(const float* __restrict__ e, float* __restrict__ enorm) {
    const int w = threadIdx.x >> 5, lane = threadIdx.x & 31;
    const int row = blockIdx.x * 8 + w;
    const float* er = e + row * C_DIM;
    float s = 0.f;
    #pragma unroll
    for (int k = lane; k < C_DIM; k += 32) { float v = er[k]; s += v * v; }
    #pragma unroll
    for (int m = 16; m >= 1; m >>= 1) s += __shfl_xor(s, m, 32);
    if (lane == 0) enorm[row] = s;
}

// ------------------------------------------- fused fp32 WMMA GEMM + argmin
// Block: 256 threads = 8 waves. Each wave owns 16 rows of z; A fragments
// (16x256 f32 = 128 VGPRs) stay register-resident across the whole code loop.
// Codebook streamed in 16-code tiles through double-buffered LDS,
// pair-interleaved so one B fragment per lane is a single 8-byte ds load.
__global__ __launch_bounds__(256) void vq_argmin(
        const float* __restrict__ z, const float* __restrict__ cb,
        const float* __restrict__ enorm,
        int* __restrict__ idx_i, float* __restrict__ idx_f) {
    __shared__ float ldsB[2][16 * C_DIM];   // 2 x 16 KB, ((k>>1)*16 + n)*2 + (k&1)

    const int tid  = threadIdx.x;
    const int w    = tid >> 5;
    const int lane = tid & 31;
    const int n    = lane & 15;          // column-within-tile / row-within-A-tile
    const int hi   = lane >> 4;          // half-wave selects K pair (A/B) / M half (C)
    const int row0 = blockIdx.x * 128;

    // ---- stage one 16-code tile: 2048 K-pairs, 8 async b64 per thread ----
    auto stage = [&](int nt, int buf) {
        #pragma unroll
        for (int j = 0; j < 8; ++j) {
            const int p    = tid + j * 256;      // pair id 0..2047
            const int code = p >> 7;             // 0..15
            const int kh   = p & 127;            // k pair 0..127  (k = 2*kh)
            const float* gsrc = cb + (size_t)(nt * 16 + code) * C_DIM + 2 * kh;
            float*       ldst = &ldsB[buf][(kh * 16 + code) * 2];
#if VQ_ASYNC
            __builtin_amdgcn_global_load_async_to_lds_b64(
                (__attribute__((address_space(1))) v2i*)gsrc,
                (__attribute__((address_space(3))) v2i*)ldst, 0, 0);
#else
            *(v2f*)ldst = *(const v2f*)gsrc;     // sync fallback
#endif
        }
    };

    // Preload this wave's A fragments for all 64 K-steps (K=4 each).
    // V_WMMA_F32_16X16X4_F32 A (16x4): M = lane&15, VGPR0/1 hold K = 2*hi, 2*hi+1.
    const float* zp = z + (size_t)(row0 + w * 16 + n) * C_DIM + 2 * hi;
    v2f a[64];
    #pragma unroll
    for (int kk = 0; kk < 64; ++kk) a[kk] = *(const v2f*)(zp + kk * 4);

    float best[8];
    int   bidx[8];
    #pragma unroll
    for (int g = 0; g < 8; ++g) { best[g] = -3.402823466e38f; bidx[g] = 0; }

    stage(0, 0);                                  // prologue prefetch

    for (int nt = 0; nt < NTILES; ++nt) {
        const int buf = nt & 1;
        // Prefetch next tile into the other buffer (it was last read in
        // iteration nt-1, whose trailing barrier already passed).
        if (nt + 1 < NTILES) {
            stage(nt + 1, buf ^ 1);
#if VQ_ASYNC
            __builtin_amdgcn_s_wait_asynccnt(8);  // only the prefetch in flight
#endif
        } else {
#if VQ_ASYNC
            __builtin_amdgcn_s_wait_asynccnt(0);
#endif
        }
        __syncthreads();                          // tile nt visible to all waves

        const float en = enorm[nt * 16 + n];

        v8f acc = {};
        #pragma unroll
        for (int kk = 0; kk < 64; ++kk) {
            // B (4x16): N = lane&15, VGPR0/1 hold K = 2*hi, 2*hi+1 (mirrors A).
            const v2f b = *(const v2f*)&ldsB[buf][((kk * 2 + hi) * 16 + n) * 2];
            acc = __builtin_amdgcn_wmma_f32_16x16x4_f32(
                      false, a[kk], false, b, (short)0, acc, false, false);
        }

        const int col = nt * 16 + n;
        #pragma unroll
        for (int g = 0; g < 8; ++g) {
            const float cand = 2.0f * acc[g] - en;   // = ||z||^2 - dist^2
            if (cand > best[g]) { best[g] = cand; bidx[g] = col; }  // strict > keeps lowest col
        }
        __syncthreads();     // all waves done reading ldsB[buf] before nt+2 overwrites it
    }

    // C/D layout: VGPR g, lanes 0-15 -> M=g, lanes 16-31 -> M=g+8; N = lane&15.
    #pragma unroll
    for (int g = 0; g < 8; ++g) {
        float v = best[g];
        int   i = bidx[g];
        #pragma unroll
        for (int msk = 1; msk < 16; msk <<= 1) {
            const float ov = __shfl_xor(v, msk, 32);
            const int   oi = __shfl_xor(i, msk, 32);
            if (ov > v || (ov == v && oi < i)) { v = ov; i = oi; }  // ties -> lowest index
        }
        if (n == 0) {                       // lanes 0 and 16 hold the two halves
            const int r = row0 + w * 16 + g + hi * 8;
            idx_i[r] = i;
            idx_f[r] = (float)i;            // d_out is float-typed
        }
    }
}

// ------------------------------------------------ gather + mask + loss sums
__global__ void vq_gather(const float* __restrict__ z, const float* __restrict__ cb,
                          const float* __restrict__ mask, const int* __restrict__ idx_i,
                          float* __restrict__ zq_out, float* __restrict__ sums) {
    const int w = threadIdx.x >> 5, lane = threadIdx.x & 31;
    const int r = blockIdx.x * 8 + w;
    const float mk = mask[r];
    const int  id  = (mk != 0.0f) ? idx_i[r] : 0;   // mask==0 -> pad with codebook[0]
    const float* er = cb + (size_t)id * C_DIM;
    const float* zr = z  + (size_t)r  * C_DIM;
    float*       o  = zq_out + (size_t)r * C_DIM;

    float local = 0.f;
    #pragma unroll
    for (int k = lane; k < C_DIM; k += 32) {
        const float q = er[k];
        o[k] = q;                            // z_q_st forward value == z_q
        const float d = zr[k] - q;
        local += d * d;
    }
    local *= mk;                             // commit term is masked
    #pragma unroll
    for (int m = 16; m >= 1; m >>= 1) local += __shfl_xor(local, m, 32);
    if (lane == 0) {
        atomicAdd(&sums[0], local);          // sum (z - z_q)^2 * mask
        atomicAdd(&sums[1], mk);             // sum mask
    }
}

__global__ void vq_init(float* sums) { if (threadIdx.x < 2) sums[threadIdx.x] = 0.f; }

__global__ void vq_finalize(const float* __restrict__ sums, float* __restrict__ out) {
    out[0] = sums[0] / (sums[1] * (float)C_DIM);
}

// ---------------------------------------------------------------- launcher
extern "C" void kernel_launch(void* const* d_in, const int* in_sizes, int n_in,
                              void* d_out, int out_size, void* d_ws, size_t ws_size,
                              hipStream_t stream) {
    const float* z    = (const float*)d_in[0];   // 16*2048*256
    const float* cb   = (const float*)d_in[1];   // 2048*256
    const float* mask = (const float*)d_in[2];   // 16*2048

    float* out   = (float*)d_out;
    float* loss  = out;                             // [0]          commitment loss
    float* zq    = out + 1;                         // [1 .. +M*C)  z_q_st
    float* idxf  = out + 1 + (size_t)M_DIM * C_DIM; // [.. +M)      indices (as float)

    float* wsf   = (float*)d_ws;                 // [0..1] loss accumulators
    float* enorm = wsf + 8;                      // 2048 floats
    int*   idxi  = (int*)(wsf + 8 + V_DIM);      // 32768 ints

    vq_init    <<<1,           32,  0, stream>>>(wsf);
    vq_enorm   <<<V_DIM / 8,   256, 0, stream>>>(cb, enorm);
    vq_argmin  <<<M_DIM / 128, 256, 0, stream>>>(z, cb, enorm, idxi, idxf);
    vq_gather  <<<M_DIM / 8,   256, 0, stream>>>(z, cb, mask, idxi, zq, wsf);
    vq_finalize<<<1,           1,   0, stream>>>(wsf, loss);
}